// Kmeans_86354612453689
// MI455X (gfx1250) — compile-verified
//
#include <hip/hip_runtime.h>
#include <math.h>

typedef __attribute__((ext_vector_type(2))) float v2f;
typedef __attribute__((ext_vector_type(8))) float v8f;

#define H 16
#define N 16384
#define K 512
#define D 128
#define EPSN 1e-6f
#define COMMIT 1e-4f

// ---------------------------------------------------------------------------
// Kernel 1: per-row stats of x: inv = 1/(||x||+eps), x2 = ||x||^2 * inv^2
// One wave (32 lanes) per (h,n) row; D=128 = 32 lanes * 4 floats.
// ---------------------------------------------------------------------------
__global__ __launch_bounds__(256) void row_stats_x(const float* __restrict__ x,
                                                   float* __restrict__ inv_arr,
                                                   float* __restrict__ x2_arr) {
  int wave = (blockIdx.x * blockDim.x + threadIdx.x) >> 5;   // row id in [0, H*N)
  int lane = threadIdx.x & 31;
  const float4 v = *(const float4*)(x + (size_t)wave * D + lane * 4);
  float s = v.x * v.x + v.y * v.y + v.z * v.z + v.w * v.w;
  #pragma unroll
  for (int off = 16; off > 0; off >>= 1) s += __shfl_xor(s, off, 32);
  if (lane == 0) {
    float norm = sqrtf(s);
    float inv  = 1.0f / (norm + EPSN);
    inv_arr[wave] = inv;
    x2_arr[wave]  = s * inv * inv;   // ||xn||^2
  }
}

// ---------------------------------------------------------------------------
// Kernel 2: per-row squared norms of means: m2[h,k] = sum_d means^2
// ---------------------------------------------------------------------------
__global__ __launch_bounds__(256) void mean_stats(const float* __restrict__ means,
                                                  float* __restrict__ m2_arr) {
  int wave = (blockIdx.x * blockDim.x + threadIdx.x) >> 5;   // row id in [0, H*K)
  int lane = threadIdx.x & 31;
  const float4 v = *(const float4*)(means + (size_t)wave * D + lane * 4);
  float s = v.x * v.x + v.y * v.y + v.z * v.z + v.w * v.w;
  #pragma unroll
  for (int off = 16; off > 0; off >>= 1) s += __shfl_xor(s, off, 32);
  if (lane == 0) m2_arr[wave] = s;
}

// ---------------------------------------------------------------------------
// Kernel 3: dists via V_WMMA_F32_16X16X4_F32.
// Each wave computes a 16(rows) x 128(clusters) strip of dists for one head.
// A = xn tile 16x4 (normalization applied on the fly), B = means^T tile 4x16.
// A layout (ISA 7.12.2, 32-bit A 16x4): lane<16: {K=0,K=1}; lane>=16: {K=2,K=3}
//   -> per-lane load of 2 consecutive floats at x[h, m0+lane&15, d0 + 2*(lane>>4)].
// B layout mirrors (rows striped, half-wave split) -> 2 consecutive floats at
//   means[h, kb + t*16 + lane&15, d0 + 2*(lane>>4)].
// ---------------------------------------------------------------------------
__global__ __launch_bounds__(256) void dists_wmma(const float* __restrict__ x,
                                                  const float* __restrict__ means,
                                                  const float* __restrict__ inv_arr,
                                                  const float* __restrict__ x2_arr,
                                                  const float* __restrict__ m2_arr,
                                                  float* __restrict__ out) {
  const int NT = N / 16;    // 1024 row tiles per head
  const int KS = K / 128;   // 4 column strips per head
  int wg   = blockIdx.x * 8 + (threadIdx.x >> 5);
  int lane = threadIdx.x & 31;

  int h   = wg / (NT * KS);
  int rem = wg % (NT * KS);
  int m0  = (rem / KS) * 16;
  int kb  = (rem % KS) * 128;

  int l15  = lane & 15;            // A row index / B-C-D column index
  int koff = (lane >> 4) << 1;     // K sub-offset within 4-wide chunk (0 or 2)

  const float* xb  = x + ((size_t)h * N + m0 + l15) * D + koff;
  float inva = inv_arr[h * N + m0 + l15];

  const float* mb[8];
  #pragma unroll
  for (int t = 0; t < 8; t++)
    mb[t] = means + ((size_t)h * K + kb + t * 16 + l15) * D + koff;

  v8f c[8] = {};

  for (int d0 = 0; d0 < D; d0 += 4) {
    float2 av = *(const float2*)(xb + d0);
    v2f a;
    a.x = av.x * inva;
    a.y = av.y * inva;
    #pragma unroll
    for (int t = 0; t < 8; t++) {
      float2 bv = *(const float2*)(mb[t] + d0);
      v2f b;
      b.x = bv.x;
      b.y = bv.y;
      c[t] = __builtin_amdgcn_wmma_f32_16x16x4_f32(
          /*neg_a=*/false, a, /*neg_b=*/false, b,
          /*c_mod=*/(short)0, c[t], /*reuse_a=*/false, /*reuse_b=*/false);
    }
  }

  // Epilogue: dist = sqrt(max(x2 + m2 - 2*dot, 0)).
  // C/D layout: VGPR v -> row M = v + 8*(lane>=16), col N = lane&15.
  int mHi = (lane >> 4) << 3;  // 0 or 8
  float x2v[8];
  #pragma unroll
  for (int v = 0; v < 8; v++) x2v[v] = x2_arr[h * N + m0 + mHi + v];

  #pragma unroll
  for (int t = 0; t < 8; t++) {
    float m2v = m2_arr[h * K + kb + t * 16 + l15];
    #pragma unroll
    for (int v = 0; v < 8; v++) {
      float d2 = x2v[v] + m2v - 2.0f * c[t][v];
      out[((size_t)h * N + m0 + mHi + v) * K + kb + t * 16 + l15] =
          sqrtf(fmaxf(d2, 0.0f));
    }
  }
}

// ---------------------------------------------------------------------------
// Kernel 4: argmin over K per (h,n) + per-row commitment-loss partial sum.
// One wave per row; deterministic (no atomics).
// ---------------------------------------------------------------------------
__global__ __launch_bounds__(256) void argmin_loss(const float* __restrict__ x,
                                                   const float* __restrict__ means,
                                                   const float* __restrict__ inv_arr,
                                                   const float* __restrict__ dists,
                                                   float* __restrict__ partials) {
  int wave = (blockIdx.x * blockDim.x + threadIdx.x) >> 5;   // row id in [0, H*N)
  int lane = threadIdx.x & 31;
  int h = wave / N;

  const float* drow = dists + (size_t)wave * K;
  float bmin = 3.402823466e38f;
  int   bidx = 0;
  for (int j = lane; j < K; j += 32) {
    float d = drow[j];
    if (d < bmin) { bmin = d; bidx = j; }
  }
  #pragma unroll
  for (int off = 16; off > 0; off >>= 1) {
    float om = __shfl_xor(bmin, off, 32);
    int   oi = __shfl_xor(bidx, off, 32);
    if (om < bmin || (om == bmin && oi < bidx)) { bmin = om; bidx = oi; }
  }
  // all lanes now agree on bidx (first-occurrence min)

  float inv = inv_arr[wave];
  const float4 xv = *(const float4*)(x + (size_t)wave * D + lane * 4);
  const float4 mv = *(const float4*)(means + ((size_t)h * K + bidx) * D + lane * 4);
  float dx = xv.x * inv - mv.x;
  float dy = xv.y * inv - mv.y;
  float dz = xv.z * inv - mv.z;
  float dw = xv.w * inv - mv.w;
  float s = dx * dx + dy * dy + dz * dz + dw * dw;
  #pragma unroll
  for (int off = 16; off > 0; off >>= 1) s += __shfl_xor(s, off, 32);
  if (lane == 0) partials[wave] = s;
}

// ---------------------------------------------------------------------------
// Kernel 5: deterministic final reduction of H*N partials -> scalar loss.
// ---------------------------------------------------------------------------
__global__ __launch_bounds__(256) void finalize_loss(const float* __restrict__ partials,
                                                     float* __restrict__ out_loss) {
  __shared__ float sm[256];
  float s = 0.0f;
  for (int i = threadIdx.x; i < H * N; i += 256) s += partials[i];
  sm[threadIdx.x] = s;
  __syncthreads();
  for (int w = 128; w > 0; w >>= 1) {
    if ((int)threadIdx.x < w) sm[threadIdx.x] += sm[threadIdx.x + w];
    __syncthreads();
  }
  if (threadIdx.x == 0)
    out_loss[0] = sm[0] * (COMMIT / ((float)H * (float)N * (float)D));
}

// ---------------------------------------------------------------------------
extern "C" void kernel_launch(void* const* d_in, const int* in_sizes, int n_in,
                              void* d_out, int out_size, void* d_ws, size_t ws_size,
                              hipStream_t stream) {
  const float* x     = (const float*)d_in[0];   // [H,N,D]
  const float* means = (const float*)d_in[1];   // [H,K,D]
  float* out = (float*)d_out;                   // [H*N*K] dists ++ [1] loss

  float* inv_arr  = (float*)d_ws;                       // H*N
  float* x2_arr   = inv_arr + (size_t)H * N;            // H*N
  float* m2_arr   = x2_arr + (size_t)H * N;             // H*K
  float* partials = m2_arr + (size_t)H * K;             // H*N

  row_stats_x<<<(H * N) / 8, 256, 0, stream>>>(x, inv_arr, x2_arr);
  mean_stats<<<(H * K) / 8, 256, 0, stream>>>(means, m2_arr);

  int tiles = H * (N / 16) * (K / 128);   // 65536 waves
  dists_wmma<<<tiles / 8, 256, 0, stream>>>(x, means, inv_arr, x2_arr, m2_arr, out);

  argmin_loss<<<(H * N) / 8, 256, 0, stream>>>(x, means, inv_arr, out, partials);
  finalize_loss<<<1, 256, 0, stream>>>(partials, out + (size_t)H * N * K);
}